// GAT_11287174054542
// MI455X (gfx1250) — compile-verified
//
#include <hip/hip_runtime.h>
#include <math.h>

// ---------------- CDNA5 (gfx1250) fp32 WMMA GAT-style fused MLP ----------------
// Node MLP (72->128->128) and all-pairs edge MLP (128->128->128->1), built on
// V_WMMA_F32_16X16X4_F32 (exact fp32 matrix math, wave32, 16x16 tiles).
// Round-2 fix: N-tile loops are rolled (#pragma unroll 1) so only ONE 8-VGPR
// accumulator is live at a time -> scheduler cannot interleave 8 acc tiles and
// spill to scratch. Inner K loop stays unrolled (16 b128 loads : 32 WMMAs).

typedef float v2f __attribute__((ext_vector_type(2)));
typedef float v4f __attribute__((ext_vector_type(4)));
typedef float v8f __attribute__((ext_vector_type(8)));

#define NB   16
#define NN   256
#define FD   128
#define NE   32640      // NN*(NN-1)/2
#define DIN  72
#define LSTR 132        // padded LDS row stride in floats (avoids bank conflicts)

// ws layout (floats)
#define WS_X    0          // 16*256*128       = 524288
#define WS_W1Q  524288     // 2*128*18*2       = 9216
#define WS_W2Q  533504     // 2*128*32*2       = 16384
#define WS_WE1Q 549888     // 2*128*32*2       = 16384
#define WS_WE2Q 566272     // 2*128*32*2       = 16384

__device__ __forceinline__ v8f wmma4(v2f a, v2f b, v8f c) {
  // D = A(16x4,f32) * B(4x16,f32) + C(16x16,f32); exact fp32 on matrix pipe
  return __builtin_amdgcn_wmma_f32_16x16x4_f32(false, a, false, b, (short)0, c,
                                               false, false);
}
__device__ __forceinline__ v2f vlo(v4f q) { return __builtin_shufflevector(q, q, 0, 1); }
__device__ __forceinline__ v2f vhi(v4f q) { return __builtin_shufflevector(q, q, 2, 3); }

// Repack W[K][128] (row-major, K = 4*S) into Wq[h][n][s] = {W[4s+2h][n],
// W[4s+2h+1][n]} so a lane's B operands for consecutive K-steps are contiguous:
// one 16B-aligned b128 load supplies two WMMAs.
__global__ void repack_kernel(const float* __restrict__ W, float* __restrict__ Wq,
                              int S) {
  int idx = blockIdx.x * blockDim.x + threadIdx.x;   // over 2*FD*S
  if (idx >= 2 * FD * S) return;
  int h   = idx / (FD * S);
  int rem = idx - h * FD * S;
  int n   = rem / S;
  int s   = rem - n * S;
  int k   = 4 * s + 2 * h;
  Wq[idx * 2 + 0] = W[k * FD + n];
  Wq[idx * 2 + 1] = W[(k + 1) * FD + n];
}

// Input feature f of node g: [type(8) | PE(60: per-octave cos(3),sin(3)) | quat(4)]
__device__ __forceinline__ float in_feat(const float* __restrict__ nt8,
                                         const float* __restrict__ pos,
                                         const float* __restrict__ quat,
                                         int g, int f) {
  if (f < 8) return nt8[g * 8 + f];
  if (f < 68) {
    int r0  = f - 8;
    int oct = r0 / 6;
    int r   = r0 - oct * 6;
    int dim = (r < 3) ? r : r - 3;
    float s = ldexpf(3.14159265358979323846f, oct);  // 2^oct * pi
    float v = s * pos[g * 3 + dim];
    return (r < 3) ? cosf(v) : sinf(v);
  }
  return quat[g * 4 + (f - 68)];
}

// ---------------- Node MLP: X = relu(relu(feat @ W1 + b1) @ W2 + b2) ----------
__global__ __launch_bounds__(128) void node_kernel(
    const float* __restrict__ nt8, const float* __restrict__ pos,
    const float* __restrict__ quat, const float* __restrict__ W1q,
    const float* __restrict__ b1, const float* __restrict__ W2q,
    const float* __restrict__ b2, float* __restrict__ X) {
  __shared__ float Hl[4][16][LSTR];
  const int wid  = threadIdx.x >> 5;
  const int lane = threadIdx.x & 31;
  const int half = lane >> 4;        // K-interleave half (A/B layouts)
  const int ln   = lane & 15;
  const int tile = blockIdx.x * 4 + wid;   // 0..255 (4096 nodes / 16)
  const int base = tile * 16;
  const int g    = base + ln;              // A-row node for this lane

  // A-matrix for layer 1: feat[g][4s + 2*half + {0,1}], s = 0..17 (K = 72)
  v2f A1[18];
#pragma unroll
  for (int s = 0; s < 18; ++s) {
    int k = 4 * s + 2 * half;
    v2f a;
    a.x = in_feat(nt8, pos, quat, g, k);
    a.y = in_feat(nt8, pos, quat, g, k + 1);
    A1[s] = a;
  }

  // Layer 1: K = 72 (18 steps); nt rolled -> one live accumulator
#pragma unroll 1
  for (int nt = 0; nt < 8; ++nt) {
    const int n = nt * 16 + ln;
    const float* Bp = W1q + (size_t)((half * FD + n) * 18) * 2;
    v8f c = {0.f, 0.f, 0.f, 0.f, 0.f, 0.f, 0.f, 0.f};
#pragma unroll
    for (int s = 0; s < 18; s += 2) {
      v4f q = *(const v4f*)(Bp + s * 2);
      c = wmma4(A1[s], vlo(q), c);
      c = wmma4(A1[s + 1], vhi(q), c);
    }
    const float bias = b1[n];
#pragma unroll
    for (int v = 0; v < 8; ++v) {
      float h = c[v] + bias;
      Hl[wid][v + 8 * half][n] = h > 0.f ? h : 0.f;
    }
  }
  __syncthreads();

  // Layer 2: K = 128 (32 steps)
  v2f A2[32];
#pragma unroll
  for (int s = 0; s < 32; ++s)
    A2[s] = *(const v2f*)(&Hl[wid][ln][4 * s + 2 * half]);

#pragma unroll 1
  for (int nt = 0; nt < 8; ++nt) {
    const int n = nt * 16 + ln;
    const float* Bp = W2q + (size_t)((half * FD + n) * 32) * 2;
    v8f c = {0.f, 0.f, 0.f, 0.f, 0.f, 0.f, 0.f, 0.f};
#pragma unroll
    for (int s = 0; s < 32; s += 2) {
      v4f q = *(const v4f*)(Bp + s * 2);
      c = wmma4(A2[s], vlo(q), c);
      c = wmma4(A2[s + 1], vhi(q), c);
    }
    const float bias = b2[n];
#pragma unroll
    for (int v = 0; v < 8; ++v) {
      float h = c[v] + bias;
      X[(size_t)(base + v + 8 * half) * FD + n] = h > 0.f ? h : 0.f;
    }
  }
}

// --------- Edge MLP: out = relu(relu((Xi+Xj)@We1+be1)@We2+be2) @ Wo + bo ------
__global__ __launch_bounds__(128) void edge_kernel(
    const float* __restrict__ X, const float* __restrict__ We1q,
    const float* __restrict__ be1, const float* __restrict__ We2q,
    const float* __restrict__ be2, const float* __restrict__ Wo,
    const float* __restrict__ bo, float* __restrict__ out) {
  __shared__ float Hl[4][16][LSTR];
  const int wid  = threadIdx.x >> 5;
  const int lane = threadIdx.x & 31;
  const int half = lane >> 4;
  const int ln   = lane & 15;
  const int tile  = blockIdx.x * 4 + wid;       // 0..32639
  const int b     = tile / (NE / 16);
  const int ebase = (tile - b * (NE / 16)) * 16;

  // decode triu edge index -> (i, j), i < j, exact via fixup
  const int e = ebase + ln;
  int i = (int)((511.0f - sqrtf(261121.0f - 8.0f * (float)e)) * 0.5f);
  i = i < 0 ? 0 : (i > 254 ? 254 : i);
  while (((i * (511 - i)) >> 1) > e) --i;
  while ((((i + 1) * (510 - i)) >> 1) <= e) ++i;
  const int j = i + 1 + (e - ((i * (511 - i)) >> 1));

  const float* xi = X + (size_t)(b * NN + i) * FD;
  const float* xj = X + (size_t)(b * NN + j) * FD;

  // A = Xi + Xj in A-layout (rows fully L2-resident: X is only 2 MB)
  v2f A[32];
#pragma unroll
  for (int s = 0; s < 32; ++s) {
    int k = 4 * s + 2 * half;
    v2f a  = *(const v2f*)(xi + k);
    v2f aj = *(const v2f*)(xj + k);
    A[s] = a + aj;
  }

  // GEMM1; nt rolled -> one live accumulator, epilogue fused
#pragma unroll 1
  for (int nt = 0; nt < 8; ++nt) {
    const int n = nt * 16 + ln;
    const float* Bp = We1q + (size_t)((half * FD + n) * 32) * 2;
    v8f c = {0.f, 0.f, 0.f, 0.f, 0.f, 0.f, 0.f, 0.f};
#pragma unroll
    for (int s = 0; s < 32; s += 2) {
      v4f q = *(const v4f*)(Bp + s * 2);
      c = wmma4(A[s], vlo(q), c);
      c = wmma4(A[s + 1], vhi(q), c);
    }
    const float bias = be1[n];
#pragma unroll
    for (int v = 0; v < 8; ++v) {
      float h = c[v] + bias;
      Hl[wid][v + 8 * half][n] = h > 0.f ? h : 0.f;
    }
  }
  __syncthreads();

  v2f A2[32];
#pragma unroll
  for (int s = 0; s < 32; ++s)
    A2[s] = *(const v2f*)(&Hl[wid][ln][4 * s + 2 * half]);

  // GEMM2; epilogue folds bias+relu+Wo-dot into 8 persistent scalars
  float p[8] = {0.f, 0.f, 0.f, 0.f, 0.f, 0.f, 0.f, 0.f};
#pragma unroll 1
  for (int nt = 0; nt < 8; ++nt) {
    const int n = nt * 16 + ln;
    const float* Bp = We2q + (size_t)((half * FD + n) * 32) * 2;
    v8f c = {0.f, 0.f, 0.f, 0.f, 0.f, 0.f, 0.f, 0.f};
#pragma unroll
    for (int s = 0; s < 32; s += 2) {
      v4f q = *(const v4f*)(Bp + s * 2);
      c = wmma4(A2[s], vlo(q), c);
      c = wmma4(A2[s + 1], vhi(q), c);
    }
    const float bias = be2[n];
    const float w    = Wo[n];
#pragma unroll
    for (int v = 0; v < 8; ++v) {
      float h = c[v] + bias;
      h = h > 0.f ? h : 0.f;
      p[v] += h * w;
    }
  }

  // D layout: each 16-lane half holds the same 8 M-rows -> butterfly within half
#pragma unroll
  for (int v = 0; v < 8; ++v) {
#pragma unroll
    for (int m = 1; m < 16; m <<= 1) p[v] += __shfl_xor(p[v], m, 16);
  }
  if (ln == 0) {
    const float b0 = bo[0];
#pragma unroll
    for (int v = 0; v < 8; ++v)
      out[(size_t)b * NE + ebase + v + 8 * half] = p[v] + b0;
  }
}

extern "C" void kernel_launch(void* const* d_in, const int* in_sizes, int n_in,
                              void* d_out, int out_size, void* d_ws, size_t ws_size,
                              hipStream_t stream) {
  (void)in_sizes; (void)n_in; (void)out_size; (void)ws_size;
  const float* node_type = (const float*)d_in[0];
  const float* pos  = (const float*)d_in[1];
  const float* quat = (const float*)d_in[2];
  const float* W1   = (const float*)d_in[3];
  const float* b1   = (const float*)d_in[4];
  const float* W2   = (const float*)d_in[5];
  const float* b2   = (const float*)d_in[6];
  const float* We1  = (const float*)d_in[7];
  const float* be1  = (const float*)d_in[8];
  const float* We2  = (const float*)d_in[9];
  const float* be2  = (const float*)d_in[10];
  const float* Wo   = (const float*)d_in[11];
  const float* bo   = (const float*)d_in[12];

  float* ws   = (float*)d_ws;
  float* Xf   = ws + WS_X;
  float* W1q  = ws + WS_W1Q;
  float* W2q  = ws + WS_W2Q;
  float* We1q = ws + WS_WE1Q;
  float* We2q = ws + WS_WE2Q;
  float* out  = (float*)d_out;

  // Repack weights into K-pair-contiguous lanes' B layout (recomputed per call).
  repack_kernel<<<(2 * FD * 18 + 255) / 256, 256, 0, stream>>>(W1, W1q, 18);
  repack_kernel<<<(2 * FD * 32 + 255) / 256, 256, 0, stream>>>(W2, W2q, 32);
  repack_kernel<<<(2 * FD * 32 + 255) / 256, 256, 0, stream>>>(We1, We1q, 32);
  repack_kernel<<<(2 * FD * 32 + 255) / 256, 256, 0, stream>>>(We2, We2q, 32);

  // 4096 nodes / 16 per wave / 4 waves per block = 64 blocks (exact)
  node_kernel<<<64, 128, 0, stream>>>(node_type, pos, quat, W1q, b1, W2q, b2, Xf);

  // 16*2040 tiles / 4 waves per block = 8160 blocks (exact)
  edge_kernel<<<(NB * (NE / 16)) / 4, 128, 0, stream>>>(Xf, We1q, be1, We2q, be2,
                                                        Wo, bo, out);
}